// FDN_37211596653125
// MI455X (gfx1250) — compile-verified
//
#include <hip/hip_runtime.h>
#include <stdint.h>

// ---------------- constants ----------------
#define IRL   44100          // IR length
#define FBINS 22051          // IRL/2 + 1
#define TLEN  441000         // samples per channel
#define ND    6              // FDN delay lines

#define TWO_PI 6.28318530717958647692f

typedef float  f2  __attribute__((ext_vector_type(2)));
typedef float  f4  __attribute__((ext_vector_type(4)));
typedef float  f8  __attribute__((ext_vector_type(8)));
typedef unsigned int u4  __attribute__((ext_vector_type(4)));
typedef unsigned int u8v __attribute__((ext_vector_type(8)));

// ============================================================
// Kernel 1: parameter prep — gamma (10x6) and U = expm(skew(U_raw)) (6x6)
// ============================================================
__device__ __forceinline__ void mm6(const float* A, const float* B, float* C) {
    for (int r = 0; r < 6; ++r)
        for (int c = 0; c < 6; ++c) {
            float s = 0.f;
            for (int k = 0; k < 6; ++k) s = fmaf(A[r*6+k], B[k*6+c], s);
            C[r*6+c] = s;
        }
}

__global__ void fdn_prep_kernel(const float* __restrict__ U_raw,
                                const float* __restrict__ gamma_raw,
                                const int*   __restrict__ delays,
                                float* __restrict__ U_ws,
                                float* __restrict__ gamma_ws) {
    const int tid = threadIdx.x;
    if (tid < 60) {                       // gamma[s][d] = sigmoid(raw)*gamma_max[d]
        int d = tid % 6;
        float df   = (float)delays[d];
        float expo = df * (-60.0f / (44100.0f * 0.75f * 20.0f));  // -60/SR/T60 * d / 20
        float gmax = __expf(expo * 2.30258509299404568402f);       // 10^expo
        float sig  = 1.0f / (1.0f + __expf(-gamma_raw[tid]));
        gamma_ws[tid] = sig * gmax;
    }
    if (tid == 0) {                       // expm via scaling & squaring + Taylor
        float A[36], E[36], Tm[36], Tn[36];
        for (int r = 0; r < 6; ++r)
            for (int c = 0; c < 6; ++c)
                A[r*6+c] = (c > r) ? U_raw[r*6+c] : ((r > c) ? -U_raw[c*6+r] : 0.f);
        float nrm = 0.f;
        for (int r = 0; r < 6; ++r) {
            float rs = 0.f;
            for (int c = 0; c < 6; ++c) rs += fabsf(A[r*6+c]);
            nrm = fmaxf(nrm, rs);
        }
        int sc = 0;
        while (nrm > 0.25f && sc < 40) { nrm *= 0.5f; ++sc; }
        float scale = ldexpf(1.0f, -sc);
        for (int k = 0; k < 36; ++k) A[k] *= scale;
        for (int k = 0; k < 36; ++k) { E[k] = A[k]; Tm[k] = A[k]; }
        for (int r = 0; r < 6; ++r) E[r*6+r] += 1.0f;            // E = I + A
        for (int it = 2; it <= 14; ++it) {                        // Taylor terms
            mm6(Tm, A, Tn);
            float invk = 1.0f / (float)it;
            for (int k = 0; k < 36; ++k) { Tm[k] = Tn[k] * invk; E[k] += Tm[k]; }
        }
        for (int q = 0; q < sc; ++q) { mm6(E, E, Tn); for (int k = 0; k < 36; ++k) E[k] = Tn[k]; }
        for (int k = 0; k < 36; ++k) U_ws[k] = E[k];
    }
}

// ============================================================
// Kernel 2: per-frequency 6x6 complex solve -> H[o][k][f]  (f2 per entry)
// ============================================================
struct cpx { float re, im; };
__device__ __forceinline__ cpx cmul(cpx a, cpx b) { return {a.re*b.re - a.im*b.im, a.re*b.im + a.im*b.re}; }
__device__ __forceinline__ cpx csub(cpx a, cpx b) { return {a.re - b.re, a.im - b.im}; }
__device__ __forceinline__ cpx cdiv(cpx a, cpx b) {
    float inv = 1.0f / (b.re*b.re + b.im*b.im);
    return {(a.re*b.re + a.im*b.im)*inv, (a.im*b.re - a.re*b.im)*inv};
}

__global__ void fdn_freq_kernel(const float* __restrict__ bvec,   // (6,2)
                                const float* __restrict__ cmat,   // (2,6)
                                const int*   __restrict__ delays,
                                const float* __restrict__ U_ws,
                                const float* __restrict__ gamma_ws,
                                float* __restrict__ Hf) {          // f2[4*FBINS]
    const int f = blockIdx.x * 256 + threadIdx.x;
    if (f >= FBINS) return;

    // interpolate gamma along decay axis (align_corners)
    float pos = (float)f * (9.0f / (float)(FBINS - 1));
    int   lo  = (int)pos; if (lo > 8) lo = 8;
    float fr  = pos - (float)lo;
    float g[6];
    for (int d = 0; d < 6; ++d)
        g[d] = gamma_ws[lo*6+d] * (1.0f - fr) + gamma_ws[lo*6+6+d] * fr;

    // M = diag(e^{j w d}) - U*g
    cpx M[6][6];
    for (int r = 0; r < 6; ++r)
        for (int c = 0; c < 6; ++c) { M[r][c].re = -U_ws[r*6+c] * g[c]; M[r][c].im = 0.f; }
    for (int r = 0; r < 6; ++r) {
        long long md = ((long long)f * (long long)delays[r]) % IRL;   // exact phase reduction
        float sn, cs; __sincosf((float)md * (TWO_PI / (float)IRL), &sn, &cs);
        M[r][r].re += cs; M[r][r].im += sn;
    }
    cpx X[6][2];
    for (int d = 0; d < 6; ++d)
        for (int k = 0; k < 2; ++k) { X[d][k].re = bvec[d*2+k]; X[d][k].im = 0.f; }

    // Gaussian elimination w/ partial pivoting, 2 RHS
    for (int col = 0; col < 6; ++col) {
        int p = col; float best = fabsf(M[col][col].re) + fabsf(M[col][col].im);
        for (int r = col + 1; r < 6; ++r) {
            float v = fabsf(M[r][col].re) + fabsf(M[r][col].im);
            if (v > best) { best = v; p = r; }
        }
        if (p != col) {
            for (int c = 0; c < 6; ++c) { cpx t = M[col][c]; M[col][c] = M[p][c]; M[p][c] = t; }
            for (int k = 0; k < 2; ++k) { cpx t = X[col][k]; X[col][k] = X[p][k]; X[p][k] = t; }
        }
        cpx piv = M[col][col];
        for (int r = col + 1; r < 6; ++r) {
            cpx fac = cdiv(M[r][col], piv);
            M[r][col].re = 0.f; M[r][col].im = 0.f;
            for (int c = col + 1; c < 6; ++c) M[r][c] = csub(M[r][c], cmul(fac, M[col][c]));
            for (int k = 0; k < 2; ++k)       X[r][k] = csub(X[r][k], cmul(fac, X[col][k]));
        }
    }
    for (int col = 5; col >= 0; --col)
        for (int k = 0; k < 2; ++k) {
            cpx a = X[col][k];
            for (int c = col + 1; c < 6; ++c) a = csub(a, cmul(M[col][c], X[c][k]));
            X[col][k] = cdiv(a, M[col][col]);
        }

    // H[o][k] = sum_i c[o][i] * X[i][k]
    f2* Ho = (f2*)Hf;
    for (int o = 0; o < 2; ++o)
        for (int k = 0; k < 2; ++k) {
            float re = 0.f, im = 0.f;
            for (int i = 0; i < 6; ++i) {
                float cc = cmat[o*6+i];
                re = fmaf(cc, X[i][k].re, re); im = fmaf(cc, X[i][k].im, im);
            }
            f2 v; v.x = re; v.y = im;
            Ho[(o*2 + k) * FBINS + f] = v;
        }
}

// ============================================================
// Kernel 3: irfft via direct DFT (exact integer phase reduction + rotation)
//   h[ch][t] = (H0 + (-1)^t H_{N/2} + 2*sum_{f=1}^{F-2} Re(H_f e^{j2pi f t/N})) / N
// ============================================================
__global__ void fdn_irfft_kernel(const float* __restrict__ Hf, float* __restrict__ h) {
    const int ch  = blockIdx.y;
    const int t   = blockIdx.x * 256 + threadIdx.x;
    __shared__ __align__(16) f2 tile[256];
    const f2* Ho = (const f2*)Hf;

    float dsn, dc;                       // rotation step cis(2*pi*t/N)
    __sincosf((float)t * (TWO_PI / (float)IRL), &dsn, &dc);
    float acc = 0.f;

    for (int f0 = 1; f0 < FBINS - 1; f0 += 256) {
        int nf = FBINS - 1 - f0; if (nf > 256) nf = 256;
        __syncthreads();
        if ((int)threadIdx.x < nf) tile[threadIdx.x] = Ho[ch * FBINS + f0 + threadIdx.x];
        __syncthreads();
        if (t < IRL) {
            long long mm = ((long long)t * (long long)f0) % IRL;     // exact reduction
            float s, c; __sincosf((float)mm * (TWO_PI / (float)IRL), &s, &c);
            for (int k = 0; k < nf; ++k) {
                f2 Hv = tile[k];
                acc = fmaf(Hv.x, c, acc);
                acc = fmaf(-Hv.y, s, acc);
                float c2 = c*dc - s*dsn; s = c*dsn + s*dc; c = c2;   // advance by one bin
            }
        }
    }
    if (t < IRL) {
        float h0 = Hf[(ch * FBINS + 0) * 2];
        float hN = Hf[(ch * FBINS + FBINS - 1) * 2];
        float sg = (t & 1) ? -1.f : 1.f;
        h[ch * IRL + t] = (h0 + sg * hN + 2.0f * acc) * (1.0f / (float)IRL);
    }
}

// ============================================================
// Kernel 4: direct convolution y[b,o,t] = sum_i (x[b,i] * h[o,i])[t]
//   via V_WMMA_F32_16X16X4_F32, x tiles staged by the Tensor Data Mover.
//   h is staged REVERSED in LDS so A-operand pairs are ascending-contiguous
//   (single ds_load_2addr_b32 in operand order, no swap movs).
// ============================================================
__device__ __forceinline__ void issue_tdm_x(const float* x, unsigned lds_off, int s0) {
    // 2D tensor view of x: 64 rows (b*2+i), stride TLEN elems; tile 64 rows x 64 cols.
    // tensor_dim0 = TLEN - s0 -> TDM zero-fills reads past end of signal.
    unsigned long long ga = (unsigned long long)(uintptr_t)x + (unsigned long long)s0 * 4ull;
    unsigned td0 = (unsigned)(TLEN - s0);
    u4 g0;
    g0[0] = 1u;                                            // count=1, user descriptor
    g0[1] = lds_off;                                       // lds_addr
    g0[2] = (unsigned)(ga & 0xFFFFFFFFull);                // global_addr[31:0]
    g0[3] = (unsigned)((ga >> 32) & 0x01FFFFFFull) | (2u << 30);  // addr[56:32] | type=2
    u8v g1;
    g1[0] = (2u << 16);                                    // data_size = 4B
    g1[1] = (td0 & 0xFFFFu) << 16;                         // tensor_dim0[15:0]
    g1[2] = (td0 >> 16) | (64u << 16);                     // tensor_dim0[31:16] | tensor_dim1 lo
    g1[3] = (64u << 16);                                   // tensor_dim1 hi=0 | tile_dim0=64
    g1[4] = 64u;                                           // tile_dim1=64, tile_dim2=0
    g1[5] = (unsigned)TLEN;                                // tensor_dim0_stride lo
    g1[6] = 0u;                                            // stride0 hi, stride1 lo
    g1[7] = 0u;
    asm volatile("tensor_load_to_lds %0, %1" :: "s"(g0), "s"(g1) : "memory");
}

__global__ __launch_bounds__(256)
void fdn_conv_kernel(const float* __restrict__ x,
                     const float* __restrict__ h,     // h[o][i][IRL]
                     float* __restrict__ y) {
    __shared__ __align__(16) float xbuf[2][64 * 64];   // [buf][row=b*2+i][kk]   32 KB
    __shared__ __align__(16) float hbuf[2][4 * 192];   // [buf][o*2+i][q] (reversed) 6 KB

    const int t0    = blockIdx.x * 128;
    const int tid   = threadIdx.x;
    const int lane  = tid & 31;
    const int wv    = tid >> 5;            // 8 waves
    const int o     = wv & 1;
    const int bt    = (wv >> 1) & 1;       // batch tile (16 batches each)
    const int mh    = (wv >> 2) & 1;       // which 64 of the 128 output times
    const int nrow  = lane & 15;           // N col (batch) / A-matrix M row
    const int khalf = lane >> 4;           // K sub-pair select (ISA A/B layout)

    f8 acc[4];
    #pragma unroll
    for (int j = 0; j < 4; ++j) acc[j] = (f8){0.f,0.f,0.f,0.f,0.f,0.f,0.f,0.f};

    int s_lo = t0 - (IRL - 1); if (s_lo < 0) s_lo = 0; s_lo &= ~63;
    int s_hi = t0 + 127; if (s_hi > TLEN - 1) s_hi = TLEN - 1; s_hi &= ~63;
    const int nchunks = ((s_hi - s_lo) >> 6) + 1;

    // ---- staging helpers ----
    // REVERSED: hbuf[buf][c][q] = h[c][dbase + 190 - q], dbase = t0 - s0 - 63, 0-padded
    #define STAGE_H(S0, BUF)                                                   \
        do {                                                                   \
            const int dbase = t0 - (S0) - 63;                                  \
            for (int slot = tid; slot < 768; slot += 256) {                    \
                int cch = slot / 192, qq = slot - cch * 192;                   \
                int d = dbase + 190 - qq;                                      \
                float v = (d >= 0 && d < IRL) ? h[cch * IRL + d] : 0.f;        \
                hbuf[BUF][cch * 192 + qq] = v;                                 \
            }                                                                  \
        } while (0)

    // prologue: chunk 0 in flight
    if (tid < 32) issue_tdm_x(x, (unsigned)(uintptr_t)&xbuf[0][0], s_lo);
    STAGE_H(s_lo, 0);

    // loop-invariant reversed-Toeplitz base: q(j, kk0) = qconst + kk0 - 16*j
    const int qconst = 127 - (mh << 6) - nrow + (khalf << 1);

    for (int c = 0; c < nchunks; ++c) {
        const int cur = c & 1, nxt = cur ^ 1;
        const int s0  = s_lo + (c << 6);

        if (c + 1 < nchunks) {                          // prefetch next chunk
            if (tid < 32) issue_tdm_x(x, (unsigned)(uintptr_t)&xbuf[nxt][0], s0 + 64);
            STAGE_H(s0 + 64, nxt);
        }
        if (tid < 32) {
            if (c + 1 < nchunks) __builtin_amdgcn_s_wait_tensorcnt(1);
            else                 __builtin_amdgcn_s_wait_tensorcnt(0);
        }
        __syncthreads();                                // xbuf[cur], hbuf[cur] ready

        // ---- GEMM: D += A(16x4 Toeplitz-h) x B(4x16 batches) ----
        #pragma unroll
        for (int i = 0; i < 2; ++i) {
            const float* hr   = &hbuf[cur][((o << 1) | i) * 192];
            const float* xrow = &xbuf[cur][(((((bt << 4) + nrow) << 1) | i) << 6) + (khalf << 1)];
            #pragma unroll 8
            for (int kk0 = 0; kk0 < 64; kk0 += 4) {
                f2 bv = *(const f2*)(xrow + kk0);        // B: VGPR0->K=kk0+2*khalf, VGPR1->+1
                const int q0 = qconst + kk0;
                f2 a0, a1, a2, a3;                       // batch A loads, then 4 wmma
                a0.x = hr[q0];      a0.y = hr[q0 + 1];
                a1.x = hr[q0 - 16]; a1.y = hr[q0 - 15];
                a2.x = hr[q0 - 32]; a2.y = hr[q0 - 31];
                a3.x = hr[q0 - 48]; a3.y = hr[q0 - 47];
                acc[0] = __builtin_amdgcn_wmma_f32_16x16x4_f32(false, a0, false, bv, (short)0, acc[0], false, false);
                acc[1] = __builtin_amdgcn_wmma_f32_16x16x4_f32(false, a1, false, bv, (short)0, acc[1], false, false);
                acc[2] = __builtin_amdgcn_wmma_f32_16x16x4_f32(false, a2, false, bv, (short)0, acc[2], false, false);
                acc[3] = __builtin_amdgcn_wmma_f32_16x16x4_f32(false, a3, false, bv, (short)0, acc[3], false, false);
            }
        }
        __syncthreads();                                 // readers done before overwrite
    }

    // ---- epilogue: C/D layout lanes0-15 -> M=r, lanes16-31 -> M=r+8 ----
    const int b = (bt << 4) + nrow;
    float* ybase = y + (size_t)((b << 1) + o) * (size_t)TLEN;
    #pragma unroll
    for (int j = 0; j < 4; ++j) {
        int tb = t0 + (mh << 6) + (j << 4) + (khalf << 3);
        if (tb + 7 < TLEN) {
            f4 v0 = {acc[j][0], acc[j][1], acc[j][2], acc[j][3]};
            f4 v1 = {acc[j][4], acc[j][5], acc[j][6], acc[j][7]};
            *(f4*)(ybase + tb)     = v0;
            *(f4*)(ybase + tb + 4) = v1;
        } else {
            for (int r = 0; r < 8; ++r)
                if (tb + r < TLEN) ybase[tb + r] = acc[j][r];
        }
    }
    #undef STAGE_H
}

// ============================================================
extern "C" void kernel_launch(void* const* d_in, const int* in_sizes, int n_in,
                              void* d_out, int out_size, void* d_ws, size_t ws_size,
                              hipStream_t stream) {
    (void)in_sizes; (void)n_in; (void)out_size; (void)ws_size;
    const float* x      = (const float*)d_in[0];   // (32, 2, 441000)
    const float* bvec   = (const float*)d_in[1];   // (6, 2)
    const float* cmat   = (const float*)d_in[2];   // (2, 6)
    const float* U_raw  = (const float*)d_in[3];   // (6, 6)
    const float* g_raw  = (const float*)d_in[4];   // (10, 6)
    const int*   delays = (const int*)d_in[5];     // (6,)
    float*       y      = (float*)d_out;           // (32, 2, 441000)

    float* ws       = (float*)d_ws;
    float* U_ws     = ws;                          // 36   (padded to 64)
    float* gamma_ws = ws + 64;                     // 60   (padded to 64)
    float* Hf_ws    = ws + 128;                    // 4 * FBINS * 2 floats
    float* h_ws     = Hf_ws + 4 * FBINS * 2;       // 4 * IRL floats

    fdn_prep_kernel<<<1, 64, 0, stream>>>(U_raw, g_raw, delays, U_ws, gamma_ws);
    fdn_freq_kernel<<<(FBINS + 255) / 256, 256, 0, stream>>>(bvec, cmat, delays,
                                                             U_ws, gamma_ws, Hf_ws);
    dim3 girf((IRL + 255) / 256, 4);
    fdn_irfft_kernel<<<girf, 256, 0, stream>>>(Hf_ws, h_ws);
    fdn_conv_kernel<<<(TLEN + 127) / 128, 256, 0, stream>>>(x, h_ws, y);
}